// GNN_global_67353677136006
// MI455X (gfx1250) — compile-verified
//
#include <hip/hip_runtime.h>

// ---------------------------------------------------------------------------
// GNN: 2x TAGConv(K=3) + leaky-relu + giant final linear, on MI455X (gfx1250)
// fp32 end-to-end; dense node-feature GEMMs via V_WMMA_F32_16X16X4_F32.
// ---------------------------------------------------------------------------

typedef __attribute__((ext_vector_type(2))) float v2f;
typedef __attribute__((ext_vector_type(8))) float v8f;

__device__ __forceinline__ void atomAddF(float* p, float v) {
    // lowers to global_atomic_add_f32 (no CAS loop)
    unsafeAtomicAdd(p, v);
}

// ---------------------------- small utility kernels ------------------------

__global__ void fill_kernel(float* __restrict__ p, float v, int n) {
    int i = blockIdx.x * blockDim.x + threadIdx.x;
    if (i < n) p[i] = v;
}

__global__ void degree_kernel(const int* __restrict__ dst, float* __restrict__ deg, int nE) {
    int e = blockIdx.x * blockDim.x + threadIdx.x;
    if (e < nE) atomAddF(&deg[dst[e]], 1.0f);
}

// deg -> deg^{-1/2} in place (0 if deg==0), matching reference gcn_norm
__global__ void dis_kernel(float* __restrict__ deg, int n) {
    int i = blockIdx.x * blockDim.x + threadIdx.x;
    if (i < n) {
        float d = deg[i];
        deg[i] = (d > 0.0f) ? rsqrtf(fmaxf(d, 1.0f)) : 0.0f;
    }
}

__global__ void norm_kernel(const int* __restrict__ src, const int* __restrict__ dst,
                            const float* __restrict__ dis, float* __restrict__ nrm, int nE) {
    int e = blockIdx.x * blockDim.x + threadIdx.x;
    if (e < nE) nrm[e] = dis[src[e]] * dis[dst[e]];
}

// out[n, d] = b[d]    (seeds the accumulator with the layer bias)
__global__ void bias_init_kernel(float* __restrict__ out, const float* __restrict__ b,
                                 int total, int doutMask) {
    int i = blockIdx.x * blockDim.x + threadIdx.x;
    if (i < total) out[i] = b[i & doutMask];
}

__global__ void leaky_kernel(float* __restrict__ x, int n) {
    int i = blockIdx.x * blockDim.x + threadIdx.x;
    if (i < n) {
        float v = x[i];
        x[i] = (v >= 0.0f) ? v : 0.01f * v;
    }
}

// -------------------------- edge propagation (scatter) ---------------------
// hout[dst] += norm_e * hin[src], feature dim D, float4 chunks per thread.
template <int D>
__global__ void scatter_kernel(const float* __restrict__ hin, float* __restrict__ hout,
                               const int* __restrict__ src, const int* __restrict__ dst,
                               const float* __restrict__ nrm, int nE) {
    constexpr int CH = D / 4;           // float4 chunks per row (8 or 16)
    int idx = blockIdx.x * blockDim.x + threadIdx.x;
    if (idx >= nE * CH) return;
    int e = idx / CH;                   // CH is a power of two -> shift
    int c = idx % CH;
    int s = src[e], d = dst[e];
    float w = nrm[e];
    const float4 hv = *(const float4*)(hin + s * D + c * 4);
    float* o = hout + d * D + c * 4;
    atomAddF(o + 0, w * hv.x);
    atomAddF(o + 1, w * hv.y);
    atomAddF(o + 2, w * hv.z);
    atomAddF(o + 3, w * hv.w);
}

// ------------------------- dense GEMM via fp32 WMMA ------------------------
// Out[N, DOUT] += H[N, DIN] @ W[DIN, DOUT]; one wave per 16x16 output tile.
// V_WMMA_F32_16X16X4_F32 fragment layouts (ISA 7.12.2):
//   A (16x4): lane%16 = M, vgpr v holds K = v + 2*(lane>=16)   -> v2f per lane
//   B (4x16): lane%16 = N, vgpr v holds K = v + 2*(lane>=16)   -> v2f per lane
//   C (16x16): vgpr r holds M = r + 8*(lane>=16), N = lane%16  -> v8f per lane
template <int DIN, int DOUT>
__global__ void wmma_accum_kernel(const float* __restrict__ H, const float* __restrict__ W,
                                  float* __restrict__ Out, int nTilesM) {
    constexpr int TN = DOUT / 16;
    int wave = (int)((blockIdx.x * blockDim.x + threadIdx.x) >> 5);
    int lane = threadIdx.x & 31;
    int numTiles = nTilesM * TN;
    if (wave >= numTiles) return;       // wave-uniform: EXEC stays all-ones
    int m0 = (wave / TN) << 4;
    int n0 = (wave % TN) << 4;
    int lm = lane & 15;
    int hi = lane >> 4;                 // 0 or 1 (lane half)

    v8f c;
#pragma unroll
    for (int r = 0; r < 8; ++r)
        c[r] = Out[(m0 + 8 * hi + r) * DOUT + n0 + lm];

#pragma unroll
    for (int k = 0; k < DIN; k += 4) {
        v2f a = *(const v2f*)(H + (m0 + lm) * DIN + k + 2 * hi);
        v2f b;
        b.x = W[(k + 2 * hi + 0) * DOUT + n0 + lm];
        b.y = W[(k + 2 * hi + 1) * DOUT + n0 + lm];
        // D = A*B + C  (fp32, round-to-nearest-even, full precision)
        c = __builtin_amdgcn_wmma_f32_16x16x4_f32(false, a, false, b,
                                                  (short)0, c, false, false);
    }

#pragma unroll
    for (int r = 0; r < 8; ++r)
        Out[(m0 + 8 * hi + r) * DOUT + n0 + lm] = c[r];
}

// ------------------------------ final linear -------------------------------
__global__ void out_init_kernel(float* __restrict__ out, const float* __restrict__ bf, int G) {
    int g = threadIdx.x;
    if (g < G) out[g] = bf[g];
}

// out[g] += sum_i h[i] * Wf[i*G + g]; threads g=0..G-1 stream contiguous rows
__global__ void gemv_kernel(const float* __restrict__ h, const float* __restrict__ Wf,
                            float* __restrict__ out, int rows, int G) {
    int g    = threadIdx.x % G;
    int rofs = threadIdx.x / G;
    int rowsPerBlk = blockDim.x / G;              // 2 for G=128, blockDim=256
    int step = rowsPerBlk * gridDim.x;
    float acc = 0.0f;
    for (int i = blockIdx.x * rowsPerBlk + rofs; i < rows; i += step) {
        __builtin_prefetch(Wf + (long long)(i + 4 * step) * G + g, 0, 1);
        acc = fmaf(h[i], Wf[i * G + g], acc);
    }
    atomAddF(&out[g], acc);
}

// ---------------------------------------------------------------------------

static inline int cdiv(int a, int b) { return (a + b - 1) / b; }

extern "C" void kernel_launch(void* const* d_in, const int* in_sizes, int n_in,
                              void* d_out, int out_size, void* d_ws, size_t ws_size,
                              hipStream_t stream) {
    const float* x  = (const float*)d_in[0];   // [N, 32]
    const int*   ei = (const int*)  d_in[1];   // [2, E]
    const float* W1 = (const float*)d_in[2];   // [4, 32, 64]
    const float* b1 = (const float*)d_in[3];   // [64]
    const float* W2 = (const float*)d_in[4];   // [4, 64, 64]
    const float* b2 = (const float*)d_in[5];   // [64]
    const float* Wf = (const float*)d_in[6];   // [64*N, G]
    const float* bf = (const float*)d_in[7];   // [G]
    float* out = (float*)d_out;

    const int N = in_sizes[0] / 32;
    const int E = in_sizes[1] / 2;
    const int G = in_sizes[7];
    const int* srcI = ei;
    const int* dstI = ei + E;

    // ---- workspace layout (floats, 1KB-aligned slabs) ----
    float* ws  = (float*)d_ws;
    size_t off = 0;
    auto alloc = [&](size_t n) { float* p = ws + off; off += (n + 255) & ~(size_t)255; return p; };
    float* deg  = alloc(N);        // degree -> deg^{-1/2} in place
    float* nrm  = alloc(E);        // per-edge gcn norm
    float* pA   = alloc((size_t)N * 64);   // propagation ping
    float* pB   = alloc((size_t)N * 64);   // propagation pong
    float* h1   = alloc((size_t)N * 64);   // layer-1 output
    float* h2   = alloc((size_t)N * 64);   // layer-2 output

    const int B = 256;
    const int tilesM = N / 16;                               // N = 20000 -> 1250
    const int wmmaBlocks = cdiv(tilesM * 4 * 32, B);         // DOUT=64 -> 4 N-tiles

    // ---- gcn_norm ----
    fill_kernel  <<<cdiv(N, B), B, 0, stream>>>(deg, 0.0f, N);
    degree_kernel<<<cdiv(E, B), B, 0, stream>>>(dstI, deg, E);
    dis_kernel   <<<cdiv(N, B), B, 0, stream>>>(deg, N);
    norm_kernel  <<<cdiv(E, B), B, 0, stream>>>(srcI, dstI, deg, nrm, E);

    // ---- layer 1: TAGConv 32 -> 64, K=3 ----
    bias_init_kernel<<<cdiv(N * 64, B), B, 0, stream>>>(h1, b1, N * 64, 63);
    wmma_accum_kernel<32, 64><<<wmmaBlocks, B, 0, stream>>>(x, W1 + 0 * 32 * 64, h1, tilesM);

    fill_kernel<<<cdiv(N * 32, B), B, 0, stream>>>(pA, 0.0f, N * 32);
    scatter_kernel<32><<<cdiv(E * 8, B), B, 0, stream>>>(x, pA, srcI, dstI, nrm, E);
    wmma_accum_kernel<32, 64><<<wmmaBlocks, B, 0, stream>>>(pA, W1 + 1 * 32 * 64, h1, tilesM);

    fill_kernel<<<cdiv(N * 32, B), B, 0, stream>>>(pB, 0.0f, N * 32);
    scatter_kernel<32><<<cdiv(E * 8, B), B, 0, stream>>>(pA, pB, srcI, dstI, nrm, E);
    wmma_accum_kernel<32, 64><<<wmmaBlocks, B, 0, stream>>>(pB, W1 + 2 * 32 * 64, h1, tilesM);

    fill_kernel<<<cdiv(N * 32, B), B, 0, stream>>>(pA, 0.0f, N * 32);
    scatter_kernel<32><<<cdiv(E * 8, B), B, 0, stream>>>(pB, pA, srcI, dstI, nrm, E);
    wmma_accum_kernel<32, 64><<<wmmaBlocks, B, 0, stream>>>(pA, W1 + 3 * 32 * 64, h1, tilesM);

    leaky_kernel<<<cdiv(N * 64, B), B, 0, stream>>>(h1, N * 64);

    // ---- layer 2: TAGConv 64 -> 64, K=3 ----
    bias_init_kernel<<<cdiv(N * 64, B), B, 0, stream>>>(h2, b2, N * 64, 63);
    wmma_accum_kernel<64, 64><<<wmmaBlocks, B, 0, stream>>>(h1, W2 + 0 * 64 * 64, h2, tilesM);

    fill_kernel<<<cdiv(N * 64, B), B, 0, stream>>>(pA, 0.0f, N * 64);
    scatter_kernel<64><<<cdiv(E * 16, B), B, 0, stream>>>(h1, pA, srcI, dstI, nrm, E);
    wmma_accum_kernel<64, 64><<<wmmaBlocks, B, 0, stream>>>(pA, W2 + 1 * 64 * 64, h2, tilesM);

    fill_kernel<<<cdiv(N * 64, B), B, 0, stream>>>(pB, 0.0f, N * 64);
    scatter_kernel<64><<<cdiv(E * 16, B), B, 0, stream>>>(pA, pB, srcI, dstI, nrm, E);
    wmma_accum_kernel<64, 64><<<wmmaBlocks, B, 0, stream>>>(pB, W2 + 2 * 64 * 64, h2, tilesM);

    fill_kernel<<<cdiv(N * 64, B), B, 0, stream>>>(pA, 0.0f, N * 64);
    scatter_kernel<64><<<cdiv(E * 16, B), B, 0, stream>>>(pB, pA, srcI, dstI, nrm, E);
    wmma_accum_kernel<64, 64><<<wmmaBlocks, B, 0, stream>>>(pA, W2 + 3 * 64 * 64, h2, tilesM);

    leaky_kernel<<<cdiv(N * 64, B), B, 0, stream>>>(h2, N * 64);

    // ---- final linear: [1, 64*N] @ [64*N, G] + bf  (HBM-bound Wf stream) ----
    out_init_kernel<<<1, ((G + 31) / 32) * 32, 0, stream>>>(out, bf, G);
    gemv_kernel<<<2048, B, 0, stream>>>(h2, Wf, out, N * 64, G);
}